// GraphSage_57312043598041
// MI455X (gfx1250) — compile-verified
//
#include <hip/hip_runtime.h>

typedef __attribute__((ext_vector_type(2))) float v2f;
typedef __attribute__((ext_vector_type(8))) float v8f;

#define FDIM 128

// ---------------- degree (computed once; depends only on dst) ----------------
__global__ void sage_deg_kernel(const long long* __restrict__ dst,
                                float* __restrict__ deg, int E) {
    int e = blockIdx.x * blockDim.x + threadIdx.x;
    if (e < E) unsafeAtomicAdd(deg + dst[e], 1.0f);
}

// ---------------- scatter-add of neighbor rows: one wave32 per edge ----------
__global__ void sage_scatter_kernel(const float* __restrict__ xin,
                                    const long long* __restrict__ src,
                                    const long long* __restrict__ dst,
                                    float* __restrict__ agg, int E) {
    int w    = (blockIdx.x * blockDim.x + threadIdx.x) >> 5;
    int lane = threadIdx.x & 31;
    if (w >= E) return;
    long long s = src[w];
    long long d = dst[w];
    const float4 v = *(const float4*)(xin + s * FDIM + lane * 4);
    float* p = agg + d * FDIM + lane * 4;
    unsafeAtomicAdd(p + 0, v.x);
    unsafeAtomicAdd(p + 1, v.y);
    unsafeAtomicAdd(p + 2, v.z);
    unsafeAtomicAdd(p + 3, v.w);
}

// ---------------- agg /= max(deg, 1) ----------------------------------------
__global__ void sage_scale_kernel(float* __restrict__ agg,
                                  const float* __restrict__ deg, int total) {
    int i = blockIdx.x * blockDim.x + threadIdx.x;
    if (i >= total) return;
    float d = deg[i >> 7];           // FDIM == 128
    agg[i] = agg[i] / fmaxf(d, 1.0f);
}

// ---------------- fused dual GEMM: out = agg@Wl^T + X@Wr^T + b (opt relu) ---
// One wave computes one 16x16 output tile with V_WMMA_F32_16X16X4_F32.
__global__ void sage_gemm_kernel(const float* __restrict__ A,   // agg  [N,128]
                                 const float* __restrict__ X,   // self [N,128]
                                 const float* __restrict__ Wl,  // [OUTC,128]
                                 const float* __restrict__ Wr,  // [OUTC,128]
                                 const float* __restrict__ bias,// [OUTC]
                                 float* __restrict__ out,       // [N,OUTC]
                                 int Nrows, int OUTC, int doRelu) {
    const int colTiles = (OUTC + 15) >> 4;
    const int w    = (blockIdx.x * blockDim.x + threadIdx.x) >> 5;
    const int lane = threadIdx.x & 31;
    const int rowTile = w / colTiles;
    const int colTile = w % colTiles;
    const int row0 = rowTile << 4;
    if (row0 >= Nrows) return;                 // wave-uniform exit
    const int col0 = colTile << 4;

    // ISA layouts (wave32):
    //  A 16x4 f32: lanes 0-15 -> M=lane, v.x=K=kb+0, v.y=K=kb+1
    //              lanes 16-31 -> M=lane-16, v.x=K=kb+2, v.y=K=kb+3
    //  B 4x16 f32: lanes 0-15 -> N=lane, v.x=K=kb+0, v.y=K=kb+1
    //              lanes 16-31 -> N=lane-16, v.x=K=kb+2, v.y=K=kb+3
    const int m    = lane & 15;
    const int hi   = lane >> 4;                // 0 or 1
    const int koff = hi * 2;

    const int rclamp = (row0 + m < Nrows) ? (row0 + m) : (Nrows - 1);
    const int hcol   = (col0 + m < OUTC) ? (col0 + m) : (OUTC - 1);

    const float* arow  = A  + (long long)rclamp * FDIM + koff;
    const float* xrow  = X  + (long long)rclamp * FDIM + koff;
    const float* wlrow = Wl + (long long)hcol   * FDIM + koff;  // B = Wl^T
    const float* wrrow = Wr + (long long)hcol   * FDIM + koff;  // B = Wr^T

    v8f acc1 = {0.f, 0.f, 0.f, 0.f, 0.f, 0.f, 0.f, 0.f};
    v8f acc2 = {0.f, 0.f, 0.f, 0.f, 0.f, 0.f, 0.f, 0.f};
    #pragma unroll
    for (int kb = 0; kb < FDIM; kb += 4) {
        v2f a1 = *(const v2f*)(arow  + kb);
        v2f b1 = *(const v2f*)(wlrow + kb);
        v2f a2 = *(const v2f*)(xrow  + kb);
        v2f b2 = *(const v2f*)(wrrow + kb);
        acc1 = __builtin_amdgcn_wmma_f32_16x16x4_f32(false, a1, false, b1,
                                                     (short)0, acc1, false, false);
        acc2 = __builtin_amdgcn_wmma_f32_16x16x4_f32(false, a2, false, b2,
                                                     (short)0, acc2, false, false);
    }

    const float bb    = bias[hcol];
    const bool  colOk = (col0 + m) < OUTC;

    // C/D layout: VGPR i, lanes 0-15 -> M=i, N=lane; lanes 16-31 -> M=i+8
    #pragma unroll
    for (int i = 0; i < 8; ++i) {
        int r = row0 + i + hi * 8;
        float v = acc1[i] + acc2[i] + bb;
        if (doRelu) v = fmaxf(v, 0.0f);
        if (colOk && r < Nrows)
            out[(long long)r * OUTC + col0 + m] = v;
    }
}

extern "C" void kernel_launch(void* const* d_in, const int* in_sizes, int n_in,
                              void* d_out, int out_size, void* d_ws, size_t ws_size,
                              hipStream_t stream) {
    const float*     x   = (const float*)d_in[0];
    const long long* ei  = (const long long*)d_in[1];   // int64 [2, E]
    const float* Wl1 = (const float*)d_in[2];
    const float* Wr1 = (const float*)d_in[3];
    const float* b1  = (const float*)d_in[4];
    const float* Wl2 = (const float*)d_in[5];
    const float* Wr2 = (const float*)d_in[6];
    const float* b2  = (const float*)d_in[7];
    const float* Wl3 = (const float*)d_in[8];
    const float* Wr3 = (const float*)d_in[9];
    const float* b3  = (const float*)d_in[10];

    const int N = in_sizes[0] / FDIM;       // 100000
    const int E = in_sizes[1] / 2;          // 3200000
    const int C = in_sizes[10];             // 6

    const long long* src = ei;
    const long long* dst = ei + E;

    // workspace layout: agg[N*128] | deg[N] | bufA[N*128] | bufB[N*128]
    float* agg  = (float*)d_ws;
    float* deg  = agg  + (size_t)N * FDIM;
    float* bufA = deg  + (size_t)N;
    float* bufB = bufA + (size_t)N * FDIM;

    const int scatterBlocks = (E + 7) / 8;            // wave per edge, 8 waves/block
    const int scaleBlocks   = (N * FDIM + 255) / 256;
    const int rowTiles      = (N + 15) / 16;
    const int tilesFull     = rowTiles * (FDIM / 16); // 128-wide output
    const int tilesLast     = rowTiles * ((C + 15) / 16);

    // degree once
    hipMemsetAsync(deg, 0, (size_t)N * sizeof(float), stream);
    sage_deg_kernel<<<(E + 255) / 256, 256, 0, stream>>>(dst, deg, E);

    // ---- layer 1: agg(x) -> bufA = relu(agg@Wl1^T + x@Wr1^T + b1)
    hipMemsetAsync(agg, 0, (size_t)N * FDIM * sizeof(float), stream);
    sage_scatter_kernel<<<scatterBlocks, 256, 0, stream>>>(x, src, dst, agg, E);
    sage_scale_kernel<<<scaleBlocks, 256, 0, stream>>>(agg, deg, N * FDIM);
    sage_gemm_kernel<<<(tilesFull + 7) / 8, 256, 0, stream>>>(
        agg, x, Wl1, Wr1, b1, bufA, N, FDIM, 1);

    // ---- layer 2: agg(bufA) -> bufB
    hipMemsetAsync(agg, 0, (size_t)N * FDIM * sizeof(float), stream);
    sage_scatter_kernel<<<scatterBlocks, 256, 0, stream>>>(bufA, src, dst, agg, E);
    sage_scale_kernel<<<scaleBlocks, 256, 0, stream>>>(agg, deg, N * FDIM);
    sage_gemm_kernel<<<(tilesFull + 7) / 8, 256, 0, stream>>>(
        agg, bufA, Wl2, Wr2, b2, bufB, N, FDIM, 1);

    // ---- layer 3: agg(bufB) -> d_out [N, 6], no relu
    hipMemsetAsync(agg, 0, (size_t)N * FDIM * sizeof(float), stream);
    sage_scatter_kernel<<<scatterBlocks, 256, 0, stream>>>(bufB, src, dst, agg, E);
    sage_scale_kernel<<<scaleBlocks, 256, 0, stream>>>(agg, deg, N * FDIM);
    sage_gemm_kernel<<<(tilesLast + 7) / 8, 256, 0, stream>>>(
        agg, bufB, Wl3, Wr3, b3, (float*)d_out, N, C, 0);
}